// GraphVariationalAutoencoder_3650722201601
// MI455X (gfx1250) — compile-verified
//
#include <hip/hip_runtime.h>

typedef __attribute__((ext_vector_type(2))) float v2f;
typedef __attribute__((ext_vector_type(4))) float v4f;
typedef __attribute__((ext_vector_type(8))) float v8f;

#define GN 1024      // nodes
#define GB 32        // batches
#define HPAD 8       // storage stride for hidden features (F <= 8)
#define XWPAD 16     // LDS stride for XW (WMMA N dim), cols >= Fout are zero

// One GCN layer: Hout = act( adj[b] @ (Hin[b] @ W) )
// mode 0: tanh activation, write Hout (stride HPAD)
// mode 1: Fout==2, col0 = x@Wmu (z_mean), col1 = x@Wlv (z_log_var);
//         fuse z = mu + eps*exp(0.5*lv); write z / zm / zlv.
__global__ __launch_bounds__(128) void gcn_layer_kernel(
    const float* __restrict__ adj,   // (GB, GN, GN)
    const float* __restrict__ Hin,   // (GB, GN, HPAD)
    const float* __restrict__ Wa,    // (Fin, Fout) or Wmu (Fin,1)
    const float* __restrict__ Wb,    // Wlv (Fin,1), mode 1 only
    float* __restrict__ Hout,        // mode 0
    const float* __restrict__ eps,   // (GB, GN), mode 1
    float* __restrict__ z, float* __restrict__ zm, float* __restrict__ zlv,
    int Fin, int Fout, int mode)
{
  __shared__ float xw[GN][XWPAD];   // 64 KB
  __shared__ float wsm[8][8];

  const int b    = blockIdx.y;
  const int tile = blockIdx.x;      // 0..15 -> 64 rows each
  const int t    = threadIdx.x;

  // Stage effective weight matrix (zero-padded 8x8) into LDS.
  if (t < 64) {
    int i = t >> 3, j = t & 7;
    float w = 0.f;
    if (i < Fin && j < Fout)
      w = (mode == 0) ? Wa[i * Fout + j] : (j == 0 ? Wa[i] : Wb[i]);
    wsm[i][j] = w;
  }
  __syncthreads();

  // XW = Hin[b] @ W  ->  LDS, padded to 16 columns (zeros beyond col 7).
  for (int m = t; m < GN; m += 128) {
    const float* hrow = Hin + ((size_t)b * GN + m) * HPAD;
    #pragma unroll
    for (int j = 0; j < 8; ++j) {
      float s = 0.f;
      for (int i = 0; i < Fin; ++i) s = fmaf(hrow[i], wsm[i][j], s);
      xw[m][j] = s;
    }
    #pragma unroll
    for (int j = 8; j < XWPAD; ++j) xw[m][j] = 0.f;
  }
  __syncthreads();

  // WMMA f32 16x16x4: each wave owns a 16-row tile of adj, K = 1024.
  const int lane = t & 31;
  const int wave = t >> 5;
  const int col  = lane & 15;       // A: row M ; B: col N ; C: col N
  const int half = lane >> 4;       // 0: K={0,1}, 1: K={2,3}
  const int koff = half << 1;
  const int rowt = tile * 64 + wave * 16;

  const float* arow = adj + ((size_t)b * GN + rowt + col) * GN + koff;

  v8f acc = {0.f, 0.f, 0.f, 0.f, 0.f, 0.f, 0.f, 0.f};
  #pragma unroll 4
  for (int m = 0; m < GN; m += 4) {
    v2f a = *(const v2f*)(arow + m);          // A[M=col][K=koff..koff+1]
    v2f bf;
    bf.x = xw[m + koff][col];                 // B[K=koff  ][N=col]
    bf.y = xw[m + koff + 1][col];             // B[K=koff+1][N=col]
    acc = __builtin_amdgcn_wmma_f32_16x16x4_f32(
        false, a, false, bf, (short)0, acc, false, false);
  }

  // C layout: VGPR r, lanes 0-15 -> (M=r, N=lane); lanes 16-31 -> (M=8+r, N=lane-16)
  const int orow = rowt + (half << 3);
  if (mode == 0) {
    if (col < Fout) {
      #pragma unroll
      for (int r = 0; r < 8; ++r)
        Hout[((size_t)b * GN + orow + r) * HPAD + col] = tanhf(acc[r]);
    }
  } else {
    // pull z_log_var (col 1 of same half) into every lane, col-0 lanes write
    float lvv[8];
    #pragma unroll
    for (int r = 0; r < 8; ++r)
      lvv[r] = __shfl(acc[r], (half << 4) | 1, 32);
    if (col == 0) {
      #pragma unroll
      for (int r = 0; r < 8; ++r) {
        size_t idx = (size_t)b * GN + orow + r;
        float mu = acc[r];
        float lv = lvv[r];
        zm[idx]  = mu;
        zlv[idx] = lv;
        z[idx]   = mu + eps[idx] * __expf(0.5f * lv);
      }
    }
  }
}

// adj_pred[b][n][m] = z[b][n] * z[b][m]; pure streaming write (134 MB), NT stores.
__global__ __launch_bounds__(256) void outer_kernel(
    const float* __restrict__ z, float* __restrict__ out)
{
  size_t gid = (size_t)blockIdx.x * blockDim.x + threadIdx.x;
  size_t e   = gid * 4;                 // flat (b, n, m), m aligned to 4
  int    m   = (int)(e & (size_t)(GN - 1));
  size_t bn  = e >> 10;                 // b*GN + n
  size_t bb  = bn >> 10;                // b
  float  zn  = z[bn];
  v4f zr = *(const v4f*)(z + (bb << 10) + m);
  v4f v;
  v.x = zn * zr.x; v.y = zn * zr.y; v.z = zn * zr.z; v.w = zn * zr.w;
  __builtin_nontemporal_store(v, (v4f*)(out + e));
}

extern "C" void kernel_launch(void* const* d_in, const int* in_sizes, int n_in,
                              void* d_out, int out_size, void* d_ws, size_t ws_size,
                              hipStream_t stream) {
  const float* X   = (const float*)d_in[0];   // (32,1024,8), stride 8 == HPAD
  const float* adj = (const float*)d_in[1];   // (32,1024,1024)
  const float* eps = (const float*)d_in[2];   // (32,1024,1)
  const float* W[6];
  for (int i = 0; i < 6; ++i) W[i] = (const float*)d_in[3 + i];
  const float* Wmu = (const float*)d_in[9];
  const float* Wlv = (const float*)d_in[10];

  float* out  = (float*)d_out;
  float* z    = out;                       // 32768
  float* zm   = out + 32768;               // 32768
  float* zlv  = out + 65536;               // 32768
  float* ap   = out + 98304;               // 33554432

  float* h0 = (float*)d_ws;                         // 1 MB
  float* h1 = h0 + (size_t)GB * GN * HPAD;          // 1 MB

  static const int widths[7] = {8, 7, 6, 5, 4, 3, 2};
  dim3 grid(16, GB), blk(128);

  const float* hin = X;
  float* bufs[2] = {h0, h1};
  for (int l = 0; l < 6; ++l) {
    float* hout = bufs[l & 1];
    gcn_layer_kernel<<<grid, blk, 0, stream>>>(
        adj, hin, W[l], nullptr, hout,
        nullptr, nullptr, nullptr, nullptr,
        widths[l], widths[l + 1], 0);
    hin = hout;
  }
  // fused mu / logvar / z layer (Fin=2, Fout=2)
  gcn_layer_kernel<<<grid, blk, 0, stream>>>(
      adj, hin, Wmu, Wlv, nullptr,
      eps, z, zm, zlv, 2, 2, 1);

  // decoder: 32*1024*1024 / 4 elems per thread / 256 threads = 32768 blocks
  outer_kernel<<<32768, 256, 0, stream>>>(z, ap);
}